// PolyConvFrame_41644002902018
// MI455X (gfx1250) — compile-verified
//
#include <hip/hip_runtime.h>
#include <math.h>

typedef float v2f __attribute__((ext_vector_type(2)));
typedef float v8f __attribute__((ext_vector_type(8)));

#define F 64
#define KORD 4

__device__ __forceinline__ void atomAddF(float* p, float v) {
    __hip_atomic_fetch_add(p, v, __ATOMIC_RELAXED, __HIP_MEMORY_SCOPE_AGENT);
}

// ---------------- scalar coefficient kernel ----------------
__global__ void k_coefs(const float* __restrict__ araw, float* __restrict__ cf,
                        float coef1, float coef2, float s2, float sc,
                        float r21, float r22, float r23,
                        float r31, float r32, float r33) {
    if (threadIdx.x == 0 && blockIdx.x == 0) {
        float al[KORD];
        for (int i = 0; i < KORD; ++i) al[i] = tanhf(araw[i]);
        cf[0] = al[0] * coef1;               // level-1 x coefficient
        cf[1] = al[0] * coef2;               // level-1 adj coefficient
        {   // L = 2 uses alphas[1], alphas[0]
            float t1 = al[1] * r21, t2 = al[1] * r22, t3 = al[1] * al[0] * r23;
            cf[2] = t1 * s2;                 // t12
            cf[3] = t1 * sc + t2;            // t22
            cf[4] = t3;                      // t3
        }
        {   // L = 3 uses alphas[2], alphas[1]
            float t1 = al[2] * r31, t2 = al[2] * r32, t3 = al[2] * al[1] * r33;
            cf[5] = t1 * s2;
            cf[6] = t1 * sc + t2;
            cf[7] = t3;
        }
    }
}

// ---------------- elementwise helpers ----------------
__global__ void k_zero4(float* __restrict__ p, long n4) {
    long i = (long)blockIdx.x * blockDim.x + threadIdx.x;
    if (i < n4) ((float4*)p)[i] = make_float4(0.f, 0.f, 0.f, 0.f);
}

__global__ void k_copy4(const float* __restrict__ s, float* __restrict__ d, long n4) {
    long i = (long)blockIdx.x * blockDim.x + threadIdx.x;
    if (i < n4) ((float4*)d)[i] = ((const float4*)s)[i];
}

// raw1 = cf[0]*x + cf[1]*y
__global__ void k_comb1(const float* __restrict__ x, const float* __restrict__ y,
                        const float* __restrict__ cf, float* __restrict__ o, long n4) {
    long i = (long)blockIdx.x * blockDim.x + threadIdx.x;
    if (i >= n4) return;
    float c0 = cf[0], c1 = cf[1];
    float4 xv = ((const float4*)x)[i];
    float4 yv = ((const float4*)y)[i];
    float4 r;
    r.x = c0 * xv.x + c1 * yv.x;  r.y = c0 * xv.y + c1 * yv.y;
    r.z = c0 * xv.z + c1 * yv.z;  r.w = c0 * xv.w + c1 * yv.w;
    ((float4*)o)[i] = r;
}

// out = cf[c+0]*y - cf[c+1]*p1 - cf[c+2]*p2
__global__ void k_combL(const float* __restrict__ y, const float* __restrict__ p1,
                        const float* __restrict__ p2, const float* __restrict__ cf,
                        int c, float* __restrict__ o, long n4) {
    long i = (long)blockIdx.x * blockDim.x + threadIdx.x;
    if (i >= n4) return;
    float t12 = cf[c], t22 = cf[c + 1], t3 = cf[c + 2];
    float4 yv = ((const float4*)y)[i];
    float4 a = ((const float4*)p1)[i];
    float4 b = ((const float4*)p2)[i];
    float4 r;
    r.x = t12 * yv.x - t22 * a.x - t3 * b.x;
    r.y = t12 * yv.y - t22 * a.y - t3 * b.y;
    r.z = t12 * yv.z - t22 * a.z - t3 * b.z;
    r.w = t12 * yv.w - t22 * a.w - t3 * b.w;
    ((float4*)o)[i] = r;
}

// ---------------- SpMV scatter: y[dst] += w * v[src] ----------------
__global__ void k_spmv(const int* __restrict__ dst, const int* __restrict__ src,
                       const float* __restrict__ w, const float* __restrict__ v,
                       float* __restrict__ y, int E) {
    long t = (long)blockIdx.x * blockDim.x + threadIdx.x;
    int e = (int)(t >> 4);
    int f4 = (int)(t & 15) * 4;
    if (e >= E) return;
    int s = src[e], d = dst[e];
    float wt = w[e];
    float4 xv = *(const float4*)(v + (size_t)s * F + f4);
    float* o = y + (size_t)d * F + f4;
    atomAddF(o + 0, wt * xv.x);
    atomAddF(o + 1, wt * xv.y);
    atomAddF(o + 2, wt * xv.z);
    atomAddF(o + 3, wt * xv.w);
}

// ---------------- column sum (for mean q) ----------------
__global__ __launch_bounds__(256)
void k_colsum(const float* __restrict__ raw, float* __restrict__ q, int nrows) {
    __shared__ float sred[256];
    int f = threadIdx.x & 63;
    int sub = threadIdx.x >> 6;           // 0..3
    int r0 = blockIdx.x * 1024;
    int r1 = r0 + 1024; if (r1 > nrows) r1 = nrows;
    float s = 0.f;
    for (int r = r0 + sub; r < r1; r += 4)
        s += raw[(size_t)r * F + f];
    sred[threadIdx.x] = s;
    __syncthreads();
    if (sub == 0) {
        float tot = sred[f] + sred[64 + f] + sred[128 + f] + sred[192 + f];
        atomAddF(q + f, tot);
    }
}

// ---------------- WMMA GEMM: C[nrows,64] = A[nrows,64] @ W^T + bias ----------------
// W is [64 out, 64 in] row-major (PyTorch Linear layout): B[f,o] = W[o*64 + f]
__global__ __launch_bounds__(256)
void k_gemm64_wmma(const float* __restrict__ A, const float* __restrict__ W,
                   const float* __restrict__ bias, float* __restrict__ C, int nrows) {
    __shared__ float ldsW[F * F];
    // stage 16KB weight into LDS
    for (int i = threadIdx.x; i < (F * F) / 4; i += 256)
        ((float4*)ldsW)[i] = ((const float4*)W)[i];
    __syncthreads();

    int lane = threadIdx.x & 31;
    int wave = threadIdx.x >> 5;                  // 0..7
    int row_tile = blockIdx.x * 2 + (wave >> 2);  // 2 row-tiles per block
    int col_tile = wave & 3;                      // 4 col-tiles cover 64 outputs
    int half = lane >> 4;                         // 0: K+0/1, 1: K+2/3
    int l = lane & 15;
    int row = row_tile * 16 + l;
    int col = col_tile * 16 + l;

    if (row_tile * 16 >= nrows) return;           // wave-uniform guard

    v8f acc = {};
    const float* arow = A + (size_t)row * F;
    const float* brow = ldsW + (size_t)col * F;
#pragma unroll
    for (int kk = 0; kk < F; kk += 4) {
        int kb = kk + half * 2;
        v2f a = *(const v2f*)(arow + kb);          // A[row, kb..kb+1]
        v2f b = *(const v2f*)(brow + kb);          // B[kb..kb+1, col] = W[col, kb..]
        acc = __builtin_amdgcn_wmma_f32_16x16x4_f32(
            false, a, false, b, (short)0, acc, false, false);
    }
    float bv = bias[col];
    float* cbase = C + (size_t)(row_tile * 16) * F + col;
#pragma unroll
    for (int j = 0; j < 8; ++j) {
        int m = j + half * 8;                      // C layout: VGPR j -> M=j (+8 for hi half)
        cbase[(size_t)m * F] = acc[j] + bv;
    }
}

// ---------------- attention logits: att[k,n] = tanh((q[k]/N) . xproj[k,n,:]) ----------------
__global__ void k_att(const float* __restrict__ xproj, const float* __restrict__ q,
                      float* __restrict__ att, int nrows, float invN) {
    int n = blockIdx.x * blockDim.x + threadIdx.x;
    int k = blockIdx.y;
    if (n >= nrows) return;
    const float* row = xproj + ((size_t)k * nrows + n) * F;
    const float* qk = q + k * F;
    float s = 0.f;
#pragma unroll
    for (int f = 0; f < F; f += 4) {
        float4 xv = *(const float4*)(row + f);
        float4 qv = *(const float4*)(qk + f);
        s += xv.x * qv.x + xv.y * qv.y + xv.z * qv.z + xv.w * qv.w;
    }
    att[(size_t)k * nrows + n] = tanhf(s * invN);
}

// ---------------- softmax over K + fuse ----------------
__global__ void k_fuse(const float* __restrict__ xproj, const float* __restrict__ att,
                       float* __restrict__ fused, int nrows) {
    long t = (long)blockIdx.x * blockDim.x + threadIdx.x;
    int n = (int)(t >> 4);
    int f = (int)(t & 15) * 4;
    if (n >= nrows) return;
    float a0 = att[n], a1 = att[(size_t)nrows + n];
    float a2 = att[2 * (size_t)nrows + n], a3 = att[3 * (size_t)nrows + n];
    float m = fmaxf(fmaxf(a0, a1), fmaxf(a2, a3));
    float e0 = expf(a0 - m), e1 = expf(a1 - m), e2 = expf(a2 - m), e3 = expf(a3 - m);
    float inv = 1.f / (e0 + e1 + e2 + e3);
    e0 *= inv; e1 *= inv; e2 *= inv; e3 *= inv;
    size_t off = (size_t)n * F + f;
    size_t kstride = (size_t)nrows * F;
    float4 x0 = *(const float4*)(xproj + off);
    float4 x1 = *(const float4*)(xproj + kstride + off);
    float4 x2 = *(const float4*)(xproj + 2 * kstride + off);
    float4 x3 = *(const float4*)(xproj + 3 * kstride + off);
    float4 r;
    r.x = e0 * x0.x + e1 * x1.x + e2 * x2.x + e3 * x3.x;
    r.y = e0 * x0.y + e1 * x1.y + e2 * x2.y + e3 * x3.y;
    r.z = e0 * x0.z + e1 * x1.z + e2 * x2.z + e3 * x3.z;
    r.w = e0 * x0.w + e1 * x1.w + e2 * x2.w + e3 * x3.w;
    *(float4*)(fused + off) = r;
}

extern "C" void kernel_launch(void* const* d_in, const int* in_sizes, int n_in,
                              void* d_out, int out_size, void* d_ws, size_t ws_size,
                              hipStream_t stream) {
    const float* x        = (const float*)d_in[0];
    const int*   ei       = (const int*)d_in[1];   // [2,E]: dst row then src row
    const float* ew       = (const float*)d_in[2];
    const float* araw     = (const float*)d_in[3];
    const float* W_att    = (const float*)d_in[4]; // [4,64,64]
    const float* b_att    = (const float*)d_in[5]; // [4,64]
    const float* W_fc     = (const float*)d_in[6]; // [64,64]
    const float* b_fc     = (const float*)d_in[7]; // [64]
    float* out = (float*)d_out;

    const int N = in_sizes[0] / F;
    const int E = in_sizes[2];
    const size_t NF = (size_t)N * F;

    float* ws    = (float*)d_ws;
    float* raw   = ws;                 // 4*NF
    float* xproj = raw + 4 * NF;       // 4*NF
    float* ybuf  = xproj + 4 * NF;     // NF   (spmv scratch, then fused)
    float* qbuf  = ybuf + NF;          // 256
    float* coefs = qbuf + 4 * F;       // 8
    float* attb  = coefs + 8;          // 4*N

    // host-side pure-constant Jacobi ratios (a=b=1, l=-1, r=1 in reference)
    const double a = 1.0, b = 1.0, l = -1.0, r = 1.0;
    float coef1 = (float)((a - b) / 2.0 - (a + b + 2.0) / 2.0 * (l + r) / (r - l));
    float coef2 = (float)((a + b + 2.0) / (r - l));
    float s2 = (float)(2.0 / (r - l));
    float sc = (float)((r + l) / (r - l));
    float rr[2][3];
    for (int L = 2; L <= 3; ++L) {
        double cl  = 2.0 * L * (L + a + b) * (2.0 * L - 2.0 + a + b);
        double c11 = (2.0 * L + a + b - 1.0) * (2.0 * L + a + b) * (2.0 * L + a + b - 2.0);
        double c12 = (2.0 * L + a + b - 1.0) * (a * a - b * b);
        double c2  = 2.0 * (L - 1.0 + a) * (L - 1.0 + b) * (2.0 * L + a + b);
        rr[L - 2][0] = (float)(c11 / cl);
        rr[L - 2][1] = (float)(c12 / cl);
        rr[L - 2][2] = (float)(c2 / cl);
    }

    const int BT = 256;
    long n4 = (long)(NF / 4);
    int gElem = (int)((n4 + BT - 1) / BT);
    int gSpmv = (int)(((long)E * 16 + BT - 1) / BT);
    int gGemm = (N / 16 + 1) / 2 + 1;          // 2 row-tiles per block
    if ((N % 32) == 0) gGemm = N / 32;

    const int* edst = ei;
    const int* esrc = ei + E;

    // 1) coefficients
    k_coefs<<<1, 64, 0, stream>>>(araw, coefs, coef1, coef2, s2, sc,
                                  rr[0][0], rr[0][1], rr[0][2],
                                  rr[1][0], rr[1][1], rr[1][2]);
    // 2) raw[0] = x
    k_copy4<<<gElem, BT, 0, stream>>>(x, raw, n4);

    // 3) level 1: y = A x ; raw1 = c0*x + c1*y
    k_zero4<<<gElem, BT, 0, stream>>>(ybuf, n4);
    k_spmv<<<gSpmv, BT, 0, stream>>>(edst, esrc, ew, x, ybuf, E);
    k_comb1<<<gElem, BT, 0, stream>>>(x, ybuf, coefs, raw + NF, n4);

    // 4) levels 2,3: three-term recurrence
    for (int L = 2; L <= 3; ++L) {
        const float* prev  = raw + (size_t)(L - 1) * NF;
        const float* prev2 = raw + (size_t)(L - 2) * NF;
        float* dst = raw + (size_t)L * NF;
        k_zero4<<<gElem, BT, 0, stream>>>(ybuf, n4);
        k_spmv<<<gSpmv, BT, 0, stream>>>(edst, esrc, ew, prev, ybuf, E);
        k_combL<<<gElem, BT, 0, stream>>>(ybuf, prev, prev2, coefs,
                                          2 + (L - 2) * 3, dst, n4);
    }

    // 5) q[k] = column sums (mean folded in later via 1/N)
    k_zero4<<<1, 64, 0, stream>>>(qbuf, (long)(4 * F / 4));
    int gMean = (N + 1023) / 1024;
    for (int k = 0; k < KORD; ++k)
        k_colsum<<<gMean, 256, 0, stream>>>(raw + (size_t)k * NF, qbuf + k * F, N);

    // 6) x_proj[k] = raw[k] @ W_att[k]^T + b_att[k]   (WMMA)
    for (int k = 0; k < KORD; ++k)
        k_gemm64_wmma<<<gGemm, 256, 0, stream>>>(raw + (size_t)k * NF,
                                                 W_att + (size_t)k * F * F,
                                                 b_att + (size_t)k * F,
                                                 xproj + (size_t)k * NF, N);

    // 7) attention logits
    dim3 gAtt((N + BT - 1) / BT, KORD);
    k_att<<<gAtt, BT, 0, stream>>>(xproj, qbuf, attb, N, 1.0f / (float)N);

    // 8) softmax over K + fuse -> ybuf
    int gFuse = (int)(((long)N * 16 + BT - 1) / BT);
    k_fuse<<<gFuse, BT, 0, stream>>>(xproj, attb, ybuf, N);

    // 9) out = fused @ W_fc^T + b_fc   (WMMA)
    k_gemm64_wmma<<<gGemm, 256, 0, stream>>>(ybuf, W_fc, b_fc, out, N);
}